// GraphGNN_11209864643247
// MI455X (gfx1250) — compile-verified
//
#include <hip/hip_runtime.h>
#include <hip/hip_bf16.h>

// ---------------------------------------------------------------------------
// GraphConv GNN (3 layers, H=20) + max/mean pool + linear readout for MI455X.
// Memory-bound: edge scatter (~1.3 GB, L2-resident atomics) dominates; node
// transforms use V_WMMA_F32_16X16X4_F32 (CDNA5 fp32 matrix pipe).
//
// Intermediate node features use stride-32 rows (20 valid + 12 zero pad):
//  - every row is 128B aligned -> b128 gather loads in the scatter kernel
//  - WMMA column tile 1 (cols 16..31) stores unconditionally into pad slots
//    (their weights/bias are zero), so the epilogue is branchless (EXEC stays
//    all-ones around the WMMA pipeline as the ISA requires).
// ---------------------------------------------------------------------------

#define NN   100000   // nodes
#define EE   3200000  // edges
#define FF   10       // input features
#define HH   20       // hidden dim
#define SS   32       // padded row stride for intermediate buffers
#define GG   512      // graphs
#define LL   2        // labels

typedef float v2f __attribute__((ext_vector_type(2)));
typedef float v8f __attribute__((ext_vector_type(8)));

// ---------------------------------------------------------------------------
__global__ __launch_bounds__(256)
void zero_f32(float* __restrict__ p, int n) {
    int i = blockIdx.x * blockDim.x + threadIdx.x;
    if (i < n) p[i] = 0.0f;
}

// ---------------------------------------------------------------------------
// Edge scatter: agg[dst] += x[src].  One thread per edge; vector gather of the
// source row (b128 when 16B-aligned, else b64), then D f32 atomics that
// resolve in the GL2 atomic units (agg is L2-resident).
template <int D, int XS>   // D = valid feature count, XS = input row stride
__global__ __launch_bounds__(256)
void scatter_add(const float* __restrict__ x, const int* __restrict__ ei,
                 float* __restrict__ agg) {
    int e = blockIdx.x * blockDim.x + threadIdx.x;
    if (e >= EE) return;
    const int src = ei[e];        // edge_index[0][e]
    const int dst = ei[EE + e];   // edge_index[1][e]
    const float* __restrict__ xr = x + (size_t)src * XS;
    float* __restrict__ ar = agg + (size_t)dst * SS;

    float v[D];
    if constexpr ((XS % 4 == 0) && (D % 4 == 0)) {
        const float4* __restrict__ p = (const float4*)xr;   // 16B-aligned rows
#pragma unroll
        for (int q = 0; q < D / 4; ++q) {
            float4 t = p[q];
            v[4 * q + 0] = t.x; v[4 * q + 1] = t.y;
            v[4 * q + 2] = t.z; v[4 * q + 3] = t.w;
        }
    } else {
        const float2* __restrict__ p = (const float2*)xr;   // 8B-aligned rows
#pragma unroll
        for (int q = 0; q < D / 2; ++q) {
            float2 t = p[q];
            v[2 * q + 0] = t.x; v[2 * q + 1] = t.y;
        }
    }
#pragma unroll
    for (int c = 0; c < D; ++c) atomicAdd(ar + c, v[c]);
}

// ---------------------------------------------------------------------------
// Node transform via WMMA fp32:
//   out = relu( [agg | xin] @ [W_rel ; W_root] + b ),   K = 2*D, Ncols = HH.
// Each wave computes one 16-row x 32-col (padded) output tile pair.
// V_WMMA_F32_16X16X4_F32 layouts (wave32):
//   A (16x4): lanes 0-15 hold M=lane  K={k0,k0+1};  lanes 16-31 M=lane-16 K={k0+2,k0+3}
//   B (4x16): vgpr0 lanes 0-15 row k0 / lanes 16-31 row k0+2; vgpr1 rows k0+1 / k0+3
//   C (16x16): vgpr v, lanes 0-15 -> M=v, N=lane; lanes 16-31 -> M=v+8, N=lane-16
template <int D, int XS>   // D = d_in, XS = row stride of xin (10 or SS)
__global__ __launch_bounds__(256)
void transform_wmma(const float* __restrict__ agg, const float* __restrict__ xin,
                    const float* __restrict__ Wrel, const float* __restrict__ bias,
                    const float* __restrict__ Wroot, float* __restrict__ out) {
    constexpr int K      = 2 * D;
    constexpr int KSTEPS = K / 4;           // D=10 -> 5, D=20 -> 10

    const int lane   = threadIdx.x & 31;
    const int cLane  = lane & 15;           // column within tile / A-matrix row
    const int hiHalf = lane >> 4;           // 0 or 1
    const int wave   = blockIdx.x * (blockDim.x >> 5) + (threadIdx.x >> 5);
    const int nwaves = gridDim.x * (blockDim.x >> 5);

    // --- B fragments: loop-invariant weights, kept in registers ---
    v2f bfrag[2][KSTEPS];
#pragma unroll
    for (int ct = 0; ct < 2; ++ct) {
        const int c = ct * 16 + cLane;
#pragma unroll
        for (int s = 0; s < KSTEPS; ++s) {
            const int ka = s * 4 + hiHalf * 2;
            const int kb = ka + 1;
            float w0 = 0.0f, w1 = 0.0f;
            if (c < HH) {   // pad columns get zero weights
                w0 = (ka < D) ? Wrel[ka * HH + c] : Wroot[(ka - D) * HH + c];
                w1 = (kb < D) ? Wrel[kb * HH + c] : Wroot[(kb - D) * HH + c];
            }
            v2f b; b.x = w0; b.y = w1;
            bfrag[ct][s] = b;
        }
    }
    float bval[2];
#pragma unroll
    for (int ct = 0; ct < 2; ++ct) {
        const int c = ct * 16 + cLane;
        bval[ct] = (c < HH) ? bias[c] : 0.0f;
    }

    constexpr int NTILES = NN / 16;         // 6250 exactly
    for (int t = wave; t < NTILES; t += nwaves) {
        const int row0   = t * 16;
        const int rowIdx = row0 + cLane;    // this lane's A-matrix row
        const float* __restrict__ aggRow = agg + (size_t)rowIdx * SS;
        const float* __restrict__ xRow   = xin + (size_t)rowIdx * XS;

        v8f acc0 = {};
        v8f acc1 = {};
#pragma unroll
        for (int s = 0; s < KSTEPS; ++s) {
            const int ka = s * 4 + hiHalf * 2;
            const int kb = ka + 1;
            v2f a;
            a.x = (ka < D) ? aggRow[ka] : xRow[ka - D];
            a.y = (kb < D) ? aggRow[kb] : xRow[kb - D];
            acc0 = __builtin_amdgcn_wmma_f32_16x16x4_f32(
                false, a, false, bfrag[0][s], (short)0, acc0, false, false);
            acc1 = __builtin_amdgcn_wmma_f32_16x16x4_f32(
                false, a, false, bfrag[1][s], (short)0, acc1, false, false);
        }

        // Branchless store: cols 20..31 land in the zero pad region.
#pragma unroll
        for (int v = 0; v < 8; ++v) {
            const int mrow = v + hiHalf * 8;
            float* __restrict__ orow = out + (size_t)(row0 + mrow) * SS;
            orow[cLane]      = fmaxf(acc0[v] + bval[0], 0.0f);
            orow[cLane + 16] = fmaxf(acc1[v] + bval[1], 0.0f);
        }
    }
}

// ---------------------------------------------------------------------------
// Segment max/sum/count pooling. h >= 0 (post-ReLU), so float max == unsigned
// max on the bit pattern with +0.0 init (also matches where(nonempty,·,0)).
__global__ __launch_bounds__(256)
void pool_kernel(const float* __restrict__ h, const int* __restrict__ batch,
                 float* __restrict__ sump, unsigned* __restrict__ maxp,
                 float* __restrict__ cnt) {
    int i = blockIdx.x * blockDim.x + threadIdx.x;
    if (i >= NN) return;
    const int g = batch[i];
    atomicAdd(&cnt[g], 1.0f);
    const float4* __restrict__ hrow = (const float4*)(h + (size_t)i * SS);
#pragma unroll
    for (int q = 0; q < HH / 4; ++q) {
        float4 t = hrow[q];
        float vv[4] = {t.x, t.y, t.z, t.w};
#pragma unroll
        for (int j = 0; j < 4; ++j) {
            const int c = 4 * q + j;
            atomicAdd(&sump[g * HH + c], vv[j]);
            atomicMax(&maxp[g * HH + c], __float_as_uint(vv[j]));
        }
    }
}

// ---------------------------------------------------------------------------
// pooled[G, 2H] @ W_lin[2H, L] + b_lin  (tiny GEMV, one thread per graph)
__global__ __launch_bounds__(128)
void readout_kernel(const float* __restrict__ sump, const unsigned* __restrict__ maxp,
                    const float* __restrict__ cnt, const float* __restrict__ Wlin,
                    const float* __restrict__ blin, float* __restrict__ out) {
    int g = blockIdx.x * blockDim.x + threadIdx.x;
    if (g >= GG) return;
    const float inv = 1.0f / fmaxf(cnt[g], 1.0f);
    float acc0 = blin[0], acc1 = blin[1];
#pragma unroll
    for (int k = 0; k < HH; ++k) {
        const float mx = __uint_as_float(maxp[g * HH + k]);
        const float mn = sump[g * HH + k] * inv;
        acc0 += mx * Wlin[k * LL + 0] + mn * Wlin[(HH + k) * LL + 0];
        acc1 += mx * Wlin[k * LL + 1] + mn * Wlin[(HH + k) * LL + 1];
    }
    out[g * LL + 0] = acc0;
    out[g * LL + 1] = acc1;
}

// ---------------------------------------------------------------------------
extern "C" void kernel_launch(void* const* d_in, const int* in_sizes, int n_in,
                              void* d_out, int out_size, void* d_ws, size_t ws_size,
                              hipStream_t stream) {
    const float* x      = (const float*)d_in[0];
    const int*   ei     = (const int*)d_in[1];
    const int*   batch  = (const int*)d_in[2];
    const float* Wrel1  = (const float*)d_in[3];
    const float* b1     = (const float*)d_in[4];
    const float* Wroot1 = (const float*)d_in[5];
    const float* Wrel2  = (const float*)d_in[6];
    const float* b2     = (const float*)d_in[7];
    const float* Wroot2 = (const float*)d_in[8];
    const float* Wrel3  = (const float*)d_in[9];
    const float* b3     = (const float*)d_in[10];
    const float* Wroot3 = (const float*)d_in[11];
    const float* Wlin   = (const float*)d_in[12];
    const float* blin   = (const float*)d_in[13];
    float* out = (float*)d_out;

    // Workspace layout (bytes); stride-32 rows => 12.8 MB per node buffer.
    char* ws = (char*)d_ws;
    const size_t BUF = (size_t)NN * SS * 4;             // 12,800,000 B
    float*    agg  = (float*)(ws + 0 * BUF);
    float*    h1   = (float*)(ws + 1 * BUF);
    float*    h2   = (float*)(ws + 2 * BUF);
    float*    sump = (float*)(ws + 3 * BUF);
    unsigned* maxp = (unsigned*)(ws + 3 * BUF + (size_t)GG * HH * 4);
    float*    cnt  = (float*)(ws + 3 * BUF + (size_t)2 * GG * HH * 4);

    const int ZB  = (NN * SS + 255) / 256;
    const int SB  = (EE + 255) / 256;
    const int NB  = (NN + 255) / 256;
    const int TB  = ((NN / 16) + 7) / 8;   // 8 waves/block, 1 tile/wave

    // ---- Layer 1: F=10 -> H=20 ----
    zero_f32<<<ZB, 256, 0, stream>>>(agg, NN * SS);
    scatter_add<FF, FF><<<SB, 256, 0, stream>>>(x, ei, agg);
    transform_wmma<FF, FF><<<TB, 256, 0, stream>>>(agg, x, Wrel1, b1, Wroot1, h1);

    // ---- Layer 2: H -> H ----
    zero_f32<<<ZB, 256, 0, stream>>>(agg, NN * SS);
    scatter_add<HH, SS><<<SB, 256, 0, stream>>>(h1, ei, agg);
    transform_wmma<HH, SS><<<TB, 256, 0, stream>>>(agg, h1, Wrel2, b2, Wroot2, h2);

    // ---- Layer 3: H -> H ----
    zero_f32<<<ZB, 256, 0, stream>>>(agg, NN * SS);
    scatter_add<HH, SS><<<SB, 256, 0, stream>>>(h2, ei, agg);
    transform_wmma<HH, SS><<<TB, 256, 0, stream>>>(agg, h2, Wrel3, b3, Wroot3, h1);

    // ---- Pooling + readout ----
    const int PZ = (2 * GG * HH + GG + 255) / 256;
    zero_f32<<<PZ, 256, 0, stream>>>(sump, 2 * GG * HH + GG);  // sump|maxp|cnt contiguous
    pool_kernel<<<NB, 256, 0, stream>>>(h1, batch, sump, maxp, cnt);
    readout_kernel<<<(GG + 127) / 128, 128, 0, stream>>>(sump, maxp, cnt, Wlin, blin, out);
}